// CustomGCN_9208409883145
// MI455X (gfx1250) — compile-verified
//
#include <hip/hip_runtime.h>
#include <hip/hip_bf16.h>

typedef __attribute__((ext_vector_type(2))) float v2f;
typedef __attribute__((ext_vector_type(8))) float v8f;

// ---------------------------------------------------------------------------
// Kernel 1: deg[i] = 1.0f  (self-loop contributes 1 to every node's degree)
// ---------------------------------------------------------------------------
__global__ void k_deg_init(float* __restrict__ deg, int n) {
    int i = blockIdx.x * blockDim.x + threadIdx.x;
    if (i < n) deg[i] = 1.0f;
}

// ---------------------------------------------------------------------------
// Kernel 2: deg[dst[e]] += 1 for every real edge (f32 atomics: exact counts)
// ---------------------------------------------------------------------------
__global__ void k_deg_count(const long long* __restrict__ dst,
                            float* __restrict__ deg, int E) {
    int i = blockIdx.x * blockDim.x + threadIdx.x;
    if (i < E) {
        long long d = dst[i];
        atomicAdd(&deg[d], 1.0f);
    }
}

// ---------------------------------------------------------------------------
// Kernel 3: dis[i] = deg > 0 ? rsqrt(deg) : 0   (in place)
// ---------------------------------------------------------------------------
__global__ void k_rsqrt(float* __restrict__ deg, int n) {
    int i = blockIdx.x * blockDim.x + threadIdx.x;
    if (i < n) {
        float d = deg[i];
        deg[i] = (d > 0.0f) ? rsqrtf(d) : 0.0f;
    }
}

// ---------------------------------------------------------------------------
// Kernel 4: h = x @ W^T via V_WMMA_F32_16X16X4_F32, and initialize
//           out = h * dis^2 + b   (self-loop message + bias)
//
// Block = 256 threads = 8 waves. Each wave computes one 16x16 tile of h:
//   2 row-tiles x 4 col-tiles per block.  W staged in LDS (16 KB),
//   each wave's 16x64 x-tile staged in LDS (4 KB/wave).
//
// WMMA f32 16x16x4 VGPR layouts (ISA 7.12.2):
//   A (16x4):  lane L holds M = L&15;  VGPR0 = K = (L<16?0:2), VGPR1 = K+1
//   B (4x16):  lane L holds N = L&15;  VGPR0 = K = (L<16?0:2), VGPR1 = K+1
//   C/D:       VGPR r: lanes 0-15 -> M=r, lanes 16-31 -> M=r+8; N = L&15
// ---------------------------------------------------------------------------
__global__ void __launch_bounds__(256)
k_gemm_selfloop(const float* __restrict__ x, const float* __restrict__ W,
                const float* __restrict__ bias, const float* __restrict__ dis,
                float* __restrict__ h, float* __restrict__ out, int n) {
    __shared__ float sW[64 * 64];        // 16 KB: full W (row-major, W[j][k])
    __shared__ float sX[8][16 * 64];     // 32 KB: per-wave 16x64 x tile

    const int tid  = threadIdx.x;
    const int lane = tid & 31;
    const int wave = tid >> 5;

    // Stage W into LDS (coalesced float4).
    {
        const float4* Wv  = (const float4*)W;
        float4*       sWv = (float4*)sW;
        for (int i = tid; i < (64 * 64) / 4; i += 256) sWv[i] = Wv[i];
    }

    const int row_tile = blockIdx.x * 2 + (wave >> 2);  // 16-row tile index
    const int col      = (wave & 3) * 16;               // output column base
    const int row_base = row_tile * 16;

    // Stage this wave's 16x64 x tile into LDS (coalesced float4).
    {
        const float4* xv  = (const float4*)(x + (size_t)row_base * 64);
        float4*       sxv = (float4*)sX[wave];
        for (int i = lane; i < (16 * 64) / 4; i += 32) sxv[i] = xv[i];
    }
    __syncthreads();

    const int m  = lane & 15;              // row (A) / col (B) within tile
    const int kb = (lane >> 4) << 1;       // 0 for lanes 0-15, 2 for 16-31

    v8f c = {};
    const float* sx = sX[wave];
    const float* sw = sW + (size_t)(col + m) * 64;   // row of W for col (col+m)

    #pragma unroll
    for (int kk = 0; kk < 16; ++kk) {      // K = 64 in steps of 4
        const int k0 = kk * 4 + kb;
        v2f a, b;
        a.x = sx[m * 64 + k0];
        a.y = sx[m * 64 + k0 + 1];
        b.x = sw[k0];
        b.y = sw[k0 + 1];
        // D = A(16x4) * B(4x16) + C   -> v_wmma_f32_16x16x4_f32
        c = __builtin_amdgcn_wmma_f32_16x16x4_f32(
                /*neg_a=*/false, a, /*neg_b=*/false, b,
                /*c_mod=*/(short)0, c, /*reuse_a=*/false, /*reuse_b=*/false);
    }

    const float bv = bias[col + m];
    #pragma unroll
    for (int r = 0; r < 8; ++r) {
        const int row = row_base + r + ((lane < 16) ? 0 : 8);
        const float hv = c[r];
        const size_t idx = (size_t)row * 64 + col + m;
        h[idx] = hv;
        const float ds = dis[row];
        out[idx] = hv * ds * ds + bv;       // self-loop message + bias
    }
}

// ---------------------------------------------------------------------------
// Kernel 5: edge aggregation, chunked per wave.
//
// Each wave takes a chunk of 32 edges.  Phase 1: lane L loads edge (e0+L)'s
// src/dst (coalesced 256B index loads) and its norm = dis[s]*dis[d]
// (32 independent 4B gathers in flight).  Phase 2: sweep the 32 edges,
// broadcasting (s, d, norm) via __shfl; for each edge the wave does one
// coalesced 256B gather of h[s] (float2/lane) and 64 coalesced
// global_atomic_add_f32 into out[d] (L2-resident: h+out = 51 MB << 192 MB L2).
// ---------------------------------------------------------------------------
__global__ void __launch_bounds__(256)
k_edge_agg(const long long* __restrict__ src, const long long* __restrict__ dst,
           const float* __restrict__ h, const float* __restrict__ dis,
           float* __restrict__ out, int E) {
    const int lane   = threadIdx.x & 31;
    const int wid    = (blockIdx.x * blockDim.x + threadIdx.x) >> 5;
    const int nwaves = (gridDim.x * blockDim.x) >> 5;
    const int nchunks = (E + 31) >> 5;

    for (int c = wid; c < nchunks; c += nwaves) {
        const int e0 = c << 5;

        // Prefetch next chunk's index lines (speculative, dropped if OOB).
        const int pe = ((c + nwaves) << 5) + lane;
        __builtin_prefetch(&src[pe], 0, 0);
        __builtin_prefetch(&dst[pe], 0, 0);

        // Phase 1: per-lane edge metadata (coalesced / parallel gathers).
        const int myE = e0 + lane;
        int   s_l = 0, d_l = 0;
        float n_l = 0.0f;
        if (myE < E) {
            s_l = (int)src[myE];
            d_l = (int)dst[myE];
            n_l = dis[s_l] * dis[d_l];
        }

        // Phase 2: wave-cooperative scatter of 32 edges.
        const int cnt = (E - e0 < 32) ? (E - e0) : 32;
        for (int j = 0; j < cnt; ++j) {
            const int   s    = __shfl(s_l, j);
            const int   d    = __shfl(d_l, j);
            const float norm = __shfl(n_l, j);

            const float2 hv = ((const float2*)(h + (size_t)s * 64))[lane];
            float* op = out + (size_t)d * 64 + 2 * lane;
            atomicAdd(op,     hv.x * norm);
            atomicAdd(op + 1, hv.y * norm);
        }
    }
}

// ---------------------------------------------------------------------------
// Launch
// ---------------------------------------------------------------------------
extern "C" void kernel_launch(void* const* d_in, const int* in_sizes, int n_in,
                              void* d_out, int out_size, void* d_ws, size_t ws_size,
                              hipStream_t stream) {
    const float*     x  = (const float*)d_in[0];       // [N,64]
    const long long* ei = (const long long*)d_in[1];   // [2,E] int64
    const float*     W  = (const float*)d_in[2];       // [64,64]
    const float*     b  = (const float*)d_in[3];       // [64]
    float*           out = (float*)d_out;              // [N,64]

    const int N = in_sizes[0] / 64;
    const int E = in_sizes[1] / 2;
    const long long* src = ei;
    const long long* dst = ei + E;

    // Scratch layout: h [N*64 f32], then deg/dis [N f32]
    float* h   = (float*)d_ws;
    float* deg = h + (size_t)N * 64;

    k_deg_init <<<(N + 255) / 256, 256, 0, stream>>>(deg, N);
    k_deg_count<<<(E + 255) / 256, 256, 0, stream>>>(dst, deg, E);
    k_rsqrt    <<<(N + 255) / 256, 256, 0, stream>>>(deg, N);

    // N = 100000 -> 6250 row-tiles of 16 -> 3125 blocks (2 row-tiles/block)
    k_gemm_selfloop<<<N / 32, 256, 0, stream>>>(x, W, b, deg, h, out, N);

    k_edge_agg<<<2048, 256, 0, stream>>>(src, dst, h, deg, out, E);
}